// WCELoss_39685497815232
// MI455X (gfx1250) — compile-verified
//
#include <hip/hip_runtime.h>
#include <cstdint>

// ---------------------------------------------------------------------------
// WCELoss (hard-negative mining) for MI455X / gfx1250.
// B=4, C=3, N = 64*256*256 pixels per image. f32 logits, i32 targets.
// Bandwidth-bound: 2 streaming passes (~268 MB each) + ~3 MB histogram work.
// ---------------------------------------------------------------------------

#define NPIX (1u << 22)   // 64*256*256 pixels per image
#define NIMG 4
#define NBIN 65536
#define PIX_PER_BLOCK 8192
#define BLOCKS_PER_IMG (NPIX / PIX_PER_BLOCK)   // 512

typedef float v2f __attribute__((ext_vector_type(2)));
typedef float v8f __attribute__((ext_vector_type(8)));

#if defined(__HIP_DEVICE_COMPILE__) && __has_builtin(__builtin_amdgcn_wmma_f32_16x16x4_f32)
#define HAVE_WMMA_F32 1
#endif

__device__ __forceinline__ float ce_3class(float a, float b, float c, int t) {
    float m   = fmaxf(a, fmaxf(b, c));
    float lse = m + __logf(__expf(a - m) + __expf(b - m) + __expf(c - m));
    float lt  = (t == 0) ? a : ((t == 1) ? b : c);
    return fmaxf(lse - lt, 0.0f);   // CE >= 0 keeps float bits order-monotonic
}

// ---------------------------------------------------------------------------
// Pass A: CE + fg sums (WMMA matrix-pipe reduction) + hi16 bg histogram
// grid = (512, 4), block = 256 (8 waves)
// ---------------------------------------------------------------------------
__global__ __launch_bounds__(256) void k_ce_hist(
        const float* __restrict__ in, const int* __restrict__ tgt,
        uint32_t* __restrict__ h1c, float* __restrict__ h1s,
        uint32_t* __restrict__ fgc, float* __restrict__ fgs)
{
    const int b   = blockIdx.y;
    const int tid = threadIdx.x;
    const float* L = in + (size_t)b * 3u * NPIX;
    const int*   T = tgt + (size_t)b * NPIX;
    uint32_t* hc = h1c + (size_t)b * NBIN;
    float*    hs = h1s + (size_t)b * NBIN;
    const size_t blockStart = (size_t)blockIdx.x * PIX_PER_BLOCK;

    v8f acc = {};
    v2f onesB; onesB.x = 1.0f; onesB.y = 1.0f;
    int cnt = 0;

    for (int it = 0; it < 8; ++it) {
        size_t base = blockStart + (size_t)it * 1024 + (size_t)tid * 4;
        float4 x0 = *(const float4*)(L + base);
        float4 x1 = *(const float4*)(L + NPIX + base);
        float4 x2 = *(const float4*)(L + 2u * NPIX + base);
        int4   tv = *(const int4*)(T + base);

        float l0[4] = {x0.x, x0.y, x0.z, x0.w};
        float l1[4] = {x1.x, x1.y, x1.z, x1.w};
        float l2[4] = {x2.x, x2.y, x2.z, x2.w};
        int   tg[4] = {tv.x, tv.y, tv.z, tv.w};

        float fsum0 = 0.0f, fsum1 = 0.0f;
#pragma unroll
        for (int j = 0; j < 4; ++j) {
            float cev = ce_3class(l0[j], l1[j], l2[j], tg[j]);
            bool  fg  = tg[j] > 0;
            float inc = fg ? cev : 0.0f;
            if (j < 2) fsum0 += inc; else fsum1 += inc;
            cnt += fg ? 1 : 0;
            if (!fg) {   // background: hi-16 radix histogram (count + value sum)
                uint32_t bits = __float_as_uint(cev);
                atomicAdd(&hc[bits >> 16], 1u);
                atomicAdd(&hs[bits >> 16], cev);
            }
        }
        // Matrix-pipe reduction: D = A * ones + C  row-sums 64 f32 per instr.
        v2f A; A.x = fsum0; A.y = fsum1;
#if defined(HAVE_WMMA_F32)
        acc = __builtin_amdgcn_wmma_f32_16x16x4_f32(false, A, false, onesB,
                                                    (short)0, acc, false, false);
#else
        acc[0] += A.x + A.y;   // fallback (should not be hit on gfx1250)
#endif
    }

    // Every column of acc holds the same row-sums; lanes 0..15 carry rows 0..7,
    // lanes 16..31 rows 8..15 of their column. Fold 8 components + partner lane.
    float colSum = 0.0f;
#pragma unroll
    for (int i = 0; i < 8; ++i) colSum += acc[i];
    float tot = colSum + __shfl_xor(colSum, 16, 32);
#pragma unroll
    for (int o = 16; o >= 1; o >>= 1) cnt += __shfl_xor(cnt, o, 32);
    if ((tid & 31) == 0) {
        atomicAdd(&fgs[b], tot);
        atomicAdd(&fgc[b], (uint32_t)cnt);
    }
}

// ---------------------------------------------------------------------------
// Kernel 2: per-image k, threshold bucket T, remainder r, exact sum above T
// grid = 4, block = 256
// ---------------------------------------------------------------------------
__global__ __launch_bounds__(256) void k_select(
        const uint32_t* __restrict__ h1c, const float* __restrict__ h1s,
        const uint32_t* __restrict__ fgc,
        uint32_t* __restrict__ Tb, uint32_t* __restrict__ rb,
        float* __restrict__ sAb, uint32_t* __restrict__ kb,
        uint32_t* __restrict__ bgn)
{
    const int b = blockIdx.x, tid = threadIdx.x;
    __shared__ uint32_t cC[256];
    __shared__ float    sC[256];
    const uint32_t* hc = h1c + (size_t)b * NBIN;
    const float*    hs = h1s + (size_t)b * NBIN;

    // chunk tid covers 256 bins, descending value order as tid ascends
    int lo = NBIN - (tid + 1) * 256;
    uint32_t c = 0; float s = 0.0f;
    for (int i = 0; i < 256; ++i) { c += hc[lo + i]; s += hs[lo + i]; }
    cC[tid] = c; sC[tid] = s;
    __syncthreads();

    if (tid == 0) {
        uint32_t fg = fgc[b];
        uint32_t bg = NPIX - fg;
        uint32_t mk = fg > 1024u ? fg : 1024u;
        uint32_t k  = bg < mk ? bg : mk;
        kb[b] = k; bgn[b] = bg;
        if (k == 0) { Tb[b] = 0xFFFFFFFFu; rb[b] = 0; sAb[b] = 0.0f; return; }
        uint32_t cum = 0; double sa = 0.0; int ts = 0;
        for (int t = 0; t < 256; ++t) {
            if (cum + cC[t] >= k) { ts = t; break; }
            cum += cC[t]; sa += (double)sC[t];
        }
        int hi = NBIN - ts * 256 - 1;
        uint32_t Tsel = 0, r = 0;
        for (int bin = hi; bin > hi - 256; --bin) {
            uint32_t cc = hc[bin];
            if (cum + cc >= k) { Tsel = (uint32_t)bin; r = k - cum; break; }
            cum += cc; sa += (double)hs[bin];
        }
        Tb[b] = Tsel; rb[b] = r; sAb[b] = (float)sa;
    }
}

// ---------------------------------------------------------------------------
// Pass C: lo16 histogram of bucket-T values. Streams tiles through LDS with
// CDNA5 async DMA loads (ASYNCcnt) instead of VGPR-bound loads.
// grid = (512, 4), block = 256
// ---------------------------------------------------------------------------
__global__ __launch_bounds__(256) void k_hist2(
        const float* __restrict__ in, const int* __restrict__ tgt,
        const uint32_t* __restrict__ Tb, uint32_t* __restrict__ h2c)
{
    const int b = blockIdx.y;
    const uint32_t Tbk = Tb[b];
    if (Tbk == 0xFFFFFFFFu) return;   // uniform per block: no bg work

    __shared__ float s0[1024], s1[1024], s2[1024];
    __shared__ int   st[1024];

    const int tid = threadIdx.x;
    const float* L = in + (size_t)b * 3u * NPIX;
    const int*   T = tgt + (size_t)b * NPIX;
    uint32_t* hc = h2c + (size_t)b * NBIN;
    const size_t blockStart = (size_t)blockIdx.x * PIX_PER_BLOCK;

    // LDS byte offsets (low 32 bits of the LDS-aperture generic address)
    uint32_t d0 = (uint32_t)(uintptr_t)&s0[tid * 4];
    uint32_t d1 = (uint32_t)(uintptr_t)&s1[tid * 4];
    uint32_t d2 = (uint32_t)(uintptr_t)&s2[tid * 4];
    uint32_t d3 = (uint32_t)(uintptr_t)&st[tid * 4];

    for (int it = 0; it < 8; ++it) {
        size_t base = blockStart + (size_t)it * 1024 + (size_t)tid * 4;
        const float* g0 = L + base;
        const float* g1 = L + NPIX + base;
        const float* g2 = L + 2u * NPIX + base;
        const int*   g3 = T + base;
        asm volatile("global_load_async_to_lds_b128 %0, %1, off"
                     :: "v"(d0), "v"(g0) : "memory");
        asm volatile("global_load_async_to_lds_b128 %0, %1, off"
                     :: "v"(d1), "v"(g1) : "memory");
        asm volatile("global_load_async_to_lds_b128 %0, %1, off"
                     :: "v"(d2), "v"(g2) : "memory");
        asm volatile("global_load_async_to_lds_b128 %0, %1, off"
                     :: "v"(d3), "v"(g3) : "memory");
        asm volatile("s_wait_asynccnt 0x0" ::: "memory");
        __syncthreads();
#pragma unroll
        for (int j = 0; j < 4; ++j) {
            int idx = tid * 4 + j;
            if (st[idx] == 0) {   // background pixel
                float cev = ce_3class(s0[idx], s1[idx], s2[idx], 0);
                uint32_t bits = __float_as_uint(cev);
                if ((bits >> 16) == Tbk)
                    atomicAdd(&hc[bits & 0xFFFFu], 1u);
            }
        }
        __syncthreads();   // protect LDS tile before next prefetch
    }
}

// ---------------------------------------------------------------------------
// Kernel 4: resolve lo16 threshold (ties are exact: bin == exact f32 value),
// assemble fg/bg terms, write scalar loss.  grid = 1, block = 256
// ---------------------------------------------------------------------------
__global__ __launch_bounds__(256) void k_final(
        const uint32_t* __restrict__ h2c, const uint32_t* __restrict__ Tb,
        const uint32_t* __restrict__ rb, const float* __restrict__ sAb,
        const uint32_t* __restrict__ kb, const uint32_t* __restrict__ bgn,
        const uint32_t* __restrict__ fgc, const float* __restrict__ fgs,
        float* __restrict__ out)
{
    const int tid = threadIdx.x;
    __shared__ uint32_t cC[256];
    __shared__ double   wC[256];
    double total = 0.0;

    for (int b = 0; b < NIMG; ++b) {
        uint32_t T = Tb[b], r = rb[b];
        const uint32_t* hc = h2c + (size_t)b * NBIN;
        int lo = NBIN - (tid + 1) * 256;
        uint32_t c = 0; double w = 0.0;
        for (int i = 0; i < 256; ++i) {
            uint32_t cc = hc[lo + i];
            if (cc) {
                c += cc;
                w += (double)cc *
                     (double)__uint_as_float((T << 16) | (uint32_t)(lo + i));
            }
        }
        cC[tid] = c; wC[tid] = w;
        __syncthreads();

        if (tid == 0) {
            double bgSum = (double)sAb[b];
            if (T != 0xFFFFFFFFu && r > 0) {
                uint32_t cum = 0; double wa = 0.0; int ts = 0;
                for (int t = 0; t < 256; ++t) {
                    if (cum + cC[t] >= r) { ts = t; break; }
                    cum += cC[t]; wa += wC[t];
                }
                int hi = NBIN - ts * 256 - 1;
                for (int bin = hi; bin > hi - 256; --bin) {
                    uint32_t cc = hc[bin];
                    double v = (double)__uint_as_float((T << 16) | (uint32_t)bin);
                    if (cum + cc >= r) { wa += (double)(r - cum) * v; break; }
                    cum += cc; wa += (double)cc * v;
                }
                bgSum += wa;
            }
            double fg_term = fgc[b] > 0 ? (double)fgs[b] / (double)fgc[b] : 0.0;
            uint32_t kk = kb[b] > 0 ? kb[b] : 1u;
            double bg_term = bgn[b] > 0 ? bgSum / (double)kk : 0.0;
            total += fg_term + bg_term;
        }
        __syncthreads();
    }
    if (tid == 0) out[0] = (float)(total * (1.0 / NIMG));
}

// ---------------------------------------------------------------------------
extern "C" void kernel_launch(void* const* d_in, const int* in_sizes, int n_in,
                              void* d_out, int out_size, void* d_ws, size_t ws_size,
                              hipStream_t stream)
{
    const float* in  = (const float*)d_in[0];
    const int*   tgt = (const int*)d_in[1];
    float*       out = (float*)d_out;

    // workspace layout
    uint8_t* ws = (uint8_t*)d_ws;
    const size_t HBYTES = (size_t)NIMG * NBIN * 4;        // 1 MiB each
    uint32_t* h1c = (uint32_t*)(ws);
    float*    h1s = (float*)   (ws + HBYTES);
    uint32_t* h2c = (uint32_t*)(ws + 2 * HBYTES);
    uint8_t*  sm  = ws + 3 * HBYTES;
    uint32_t* fgc = (uint32_t*)(sm + 0);
    float*    fgs = (float*)   (sm + 16);
    uint32_t* Tb  = (uint32_t*)(sm + 32);
    uint32_t* rb  = (uint32_t*)(sm + 48);
    float*    sAb = (float*)   (sm + 64);
    uint32_t* kb  = (uint32_t*)(sm + 80);
    uint32_t* bgn = (uint32_t*)(sm + 96);

    hipMemsetAsync(d_ws, 0, 3 * HBYTES + 128, stream);

    dim3 blk(256);
    dim3 gBig(BLOCKS_PER_IMG, NIMG);
    k_ce_hist<<<gBig, blk, 0, stream>>>(in, tgt, h1c, h1s, fgc, fgs);
    k_select<<<dim3(NIMG), blk, 0, stream>>>(h1c, h1s, fgc, Tb, rb, sAb, kb, bgn);
    k_hist2<<<gBig, blk, 0, stream>>>(in, tgt, Tb, h2c);
    k_final<<<dim3(1), blk, 0, stream>>>(h2c, Tb, rb, sAb, kb, bgn, fgc, fgs, out);

    (void)in_sizes; (void)n_in; (void)out_size; (void)ws_size;
}